// DaGMM_46188078301719
// MI455X (gfx1250) — compile-verified
//
#include <hip/hip_runtime.h>
#include <hip/hip_bf16.h>
#include <math.h>

// ---------------------------------------------------------------------------
// DaGMM energy for MI455X (gfx1250, wave32).
// Big contractions (weighted covariance SYRK, Mahalanobis GEMM) run on
// V_WMMA_F32_16X16X4_F32; small 66x66 Cholesky/inverse done per-k in LDS.
// ---------------------------------------------------------------------------

typedef __attribute__((ext_vector_type(2))) float v2f;
typedef __attribute__((ext_vector_type(8))) float v8f;

// Check only during the device pass (host pass cannot see amdgcn builtins).
#if defined(__HIP_DEVICE_COMPILE__) && !__has_builtin(__builtin_amdgcn_wmma_f32_16x16x4_f32)
#error "missing __builtin_amdgcn_wmma_f32_16x16x4_f32 on device"
#endif

#define NS 524288
#define KK 4
#define DD 66
#define DP 80      // D padded to 5 tiles of 16
#define DI 68      // padded row count for Sigma^-1 (multiple of 4)
#define TWO_PI 6.283185307179586f
#define EPSC 1e-6f

// workspace layout (float offsets)
#define OFF_SUMG  0
#define OFF_GZ    4
#define OFF_MU    (OFF_GZ + KK*DP)        // 324
#define OFF_COV   (OFF_MU + KK*DP)        // 644   [K][80][80] accum
#define OFF_CINV  (OFF_COV + KK*DP*DP)    // 26244 [K][68][80]
#define OFF_COEF  (OFF_CINV + KK*DI*DP)   // 48004
#define OFF_CDIAG (OFF_COEF + KK)         // 48008
#define OFF_EN    (OFF_CDIAG + 1)         // 48009
#define WS_FLOATS 48016                   // ~192 KB

// ---------------------------------------------------------------------------
__global__ __launch_bounds__(256) void k_zero(float* __restrict__ ws) {
  int i = blockIdx.x * 256 + threadIdx.x;
  if (i < WS_FLOATS) ws[i] = 0.f;
}

// ---------------------------------------------------------------------------
// Pass 1: sum_gamma[K] and gz[K,D] = sum_n gamma[n,k] z[n,d]
#define T1 64
__global__ __launch_bounds__(256) void k_pass1(const float* __restrict__ z,
                                               const float* __restrict__ g,
                                               float* __restrict__ ws) {
  __shared__ float zt[T1][DD];
  __shared__ float gt[T1][KK];
  const int tid = threadIdx.x;
  // each thread owns up to two (k, col) pairs; col==66 is the "ones" column
  const int p0 = tid, p1 = tid + 256;
  const int k0 = p0 / 67, c0 = p0 % 67;
  const int k1 = p1 / 67, c1 = p1 % 67;
  float acc0 = 0.f, acc1 = 0.f;
  const int ntiles = NS / T1;
  for (int t = blockIdx.x; t < ntiles; t += gridDim.x) {
    __syncthreads();
    const long base = (long)t * T1;
    for (int i = tid; i < T1 * DD; i += 256) ((float*)zt)[i] = z[base * DD + i];
    for (int i = tid; i < T1 * KK; i += 256) ((float*)gt)[i] = g[base * KK + i];
    __syncthreads();
    for (int r = 0; r < T1; r++) {
      acc0 += gt[r][k0] * (c0 < DD ? zt[r][c0] : 1.0f);
      if (p1 < KK * 67) acc1 += gt[r][k1] * (c1 < DD ? zt[r][c1] : 1.0f);
    }
  }
  if (c0 < DD) atomicAdd(&ws[OFF_GZ + k0 * DP + c0], acc0);
  else         atomicAdd(&ws[OFF_SUMG + k0], acc0);
  if (p1 < KK * 67) {
    if (c1 < DD) atomicAdd(&ws[OFF_GZ + k1 * DP + c1], acc1);
    else         atomicAdd(&ws[OFF_SUMG + k1], acc1);
  }
}

// ---------------------------------------------------------------------------
__global__ __launch_bounds__(320) void k_mu(float* __restrict__ ws) {
  int i = threadIdx.x;               // exactly 320 threads = K*DP
  int k = i / DP, d = i % DP;
  float m = 0.f;
  if (d < DD) m = ws[OFF_GZ + i] / ws[OFF_SUMG + k];
  ws[OFF_MU + i] = m;                // zero-padded beyond D=66
}

// ---------------------------------------------------------------------------
// Pass 2: cov accum S[k,d,e] = sum_n gamma[n,k] (z-mu)_d (z-mu)_e  via WMMA.
// 640 threads = 20 waves: wave -> (k = w/5, row-tile ti = w%5), 5 col-tile
// accumulators each. Contraction dim (samples) advances 4 per WMMA.
#define T2 64
__global__ __launch_bounds__(640) void k_cov(const float* __restrict__ z,
                                             const float* __restrict__ g,
                                             float* __restrict__ ws) {
  __shared__ float zt[T2][DP];
  __shared__ float gt[T2][KK];
  const int tid = threadIdx.x;
  const int wave = tid >> 5, lane = tid & 31;
  const int c = lane & 15, half = lane >> 4;
  const int k = wave / 5, ti = wave % 5;
  const float muA = ws[OFF_MU + k * DP + ti * 16 + c];
  float muB[5];
#pragma unroll
  for (int j = 0; j < 5; j++) muB[j] = ws[OFF_MU + k * DP + j * 16 + c];
  v8f acc[5] = {};
  const int ntiles = NS / T2;
  for (int t = blockIdx.x; t < ntiles; t += gridDim.x) {
    __syncthreads();
    const long base = (long)t * T2;
    for (int i = tid; i < T2 * DD; i += 640) {
      int r = i / DD, cc = i % DD;
      zt[r][cc] = z[base * DD + i];
    }
    for (int i = tid; i < T2 * (DP - DD); i += 640) {
      int r = i / (DP - DD), cc = DD + i % (DP - DD);
      zt[r][cc] = 0.f;
    }
    for (int i = tid; i < T2 * KK; i += 640) ((float*)gt)[i] = g[base * KK + i];
    __syncthreads();
    for (int n0 = 0; n0 < T2; n0 += 4) {
      const int s0 = n0 + 2 * half;            // contraction k = v + 2*half
      v2f a;
      a.x = gt[s0][k]     * (zt[s0][ti * 16 + c]     - muA);
      a.y = gt[s0 + 1][k] * (zt[s0 + 1][ti * 16 + c] - muA);
#pragma unroll
      for (int j = 0; j < 5; j++) {
        v2f b;
        b.x = zt[s0][j * 16 + c]     - muB[j];
        b.y = zt[s0 + 1][j * 16 + c] - muB[j];
        acc[j] = __builtin_amdgcn_wmma_f32_16x16x4_f32(
            false, a, false, b, (short)0, acc[j], false, false);
      }
    }
  }
#pragma unroll
  for (int j = 0; j < 5; j++)
#pragma unroll
    for (int v = 0; v < 8; v++) {
      int d = ti * 16 + v + 8 * half;          // C row = v + 8*half
      atomicAdd(&ws[OFF_COV + (k * DP + d) * DP + j * 16 + c], acc[j][v]);
    }
}

// ---------------------------------------------------------------------------
// Per-k: normalize cov, cov_diag, Cholesky, coef = phi/sqrt(det_cov),
// Sigma^-1 = L^-T L^-1 written zero-padded to [68][80].
__global__ __launch_bounds__(128) void k_chol(float* __restrict__ ws) {
  const int k = blockIdx.x, tid = threadIdx.x;
  __shared__ float A[DD][DD + 1];
  __shared__ float Li[DD][DD + 1];
  __shared__ float red[128];
  const float sg = ws[OFF_SUMG + k];
  for (int i = tid; i < DD * DD; i += 128) {
    int d = i / DD, e = i % DD;
    A[d][e] = ws[OFF_COV + (k * DP + d) * DP + e] / sg;
  }
  __syncthreads();
  float cd = 0.f;
  for (int d = tid; d < DD; d += 128) cd += 1.0f / A[d][d];
  red[tid] = cd;
  __syncthreads();
  for (int s = 64; s > 0; s >>= 1) { if (tid < s) red[tid] += red[tid + s]; __syncthreads(); }
  if (tid == 0) atomicAdd(&ws[OFF_CDIAG], red[0]);
  // right-looking Cholesky, lower triangle in place
  for (int j = 0; j < DD; j++) {
    if (tid == 0) A[j][j] = sqrtf(A[j][j]);
    __syncthreads();
    for (int i = j + 1 + tid; i < DD; i += 128) A[i][j] /= A[j][j];
    __syncthreads();
    int w = DD - 1 - j;
    for (int idx = tid; idx < w * w; idx += 128) {
      int ii = j + 1 + idx / w, pp = j + 1 + idx % w;
      if (pp <= ii) A[ii][pp] -= A[ii][j] * A[pp][j];
    }
    __syncthreads();
  }
  if (tid == 0) {
    float ld = 0.f;
    for (int j = 0; j < DD; j++) ld += logf(A[j][j]);
    ld += 0.5f * (float)DD * logf(TWO_PI);   // log det_cov = log sqrt(det(2pi C))
    float phi = sg / (float)NS;
    ws[OFF_COEF + k] = phi * expf(-0.5f * ld);  // phi / sqrt(det_cov)
  }
  // L^-1, column-parallel
  for (int j = tid; j < DD; j += 128) {
    Li[j][j] = 1.0f / A[j][j];
    for (int i = j + 1; i < DD; i++) {
      float s = 0.f;
      for (int p = j; p < i; p++) s += A[i][p] * Li[p][j];
      Li[i][j] = -s / A[i][i];
    }
  }
  __syncthreads();
  for (int idx = tid; idx < DI * DP; idx += 128) {
    int d = idx / DP, e = idx % DP;
    float s = 0.f;
    if (d < DD && e < DD) {
      int lo = d > e ? d : e;
      for (int i = lo; i < DD; i++) s += Li[i][d] * Li[i][e];
    }
    ws[OFF_CINV + k * DI * DP + idx] = s;    // zero-padded
  }
}

// ---------------------------------------------------------------------------
// Pass 3: quad[n,k] = zm^T Sigma^-1 zm via WMMA, then log-sum-exp energy.
// Sigma^-1 for all k lives in dynamic LDS; each wave owns a 16-sample tile.
#define LDS3_FLOATS (KK * DI * DP + 8 * 16 * DP + KK * DP + 8 * 16 * KK)
__global__ __launch_bounds__(256) void k_energy(const float* __restrict__ z,
                                                float* __restrict__ ws) {
  extern __shared__ float smem[];
  float* cinvs = smem;                        // [K][68][80]
  float* zts   = cinvs + KK * DI * DP;        // [8 waves][16][80]
  float* mus   = zts + 8 * 16 * DP;           // [K][80]
  float* qls   = mus + KK * DP;               // [8 waves][16][K]
  const int tid = threadIdx.x;
  for (int i = tid; i < KK * DI * DP; i += 256) cinvs[i] = ws[OFF_CINV + i];
  for (int i = tid; i < KK * DP; i += 256)      mus[i]   = ws[OFF_MU + i];
  __syncthreads();
  const int wave = tid >> 5, lane = tid & 31, c = lane & 15, half = lane >> 4;
  float cf[KK];
#pragma unroll
  for (int k = 0; k < KK; k++) cf[k] = ws[OFF_COEF + k];
  float* zw = zts + wave * 16 * DP;
  float esum = 0.f;
  const int ntiles = NS / 16;                  // 32768; grid 512*8 waves -> 8 each
  for (int t = blockIdx.x * 8 + wave; t < ntiles; t += gridDim.x * 8) {
    const long base = (long)t * 16;
    for (int i = lane; i < 16 * DD; i += 32) {
      int r = i / DD, cc = i % DD;
      zw[r * DP + cc] = z[base * DD + i];
    }
    for (int i = lane; i < 16 * (DP - DD); i += 32) {
      int r = i / (DP - DD), cc = DD + i % (DP - DD);
      zw[r * DP + cc] = 0.f;
    }
    asm volatile("s_wait_dscnt 0" ::: "memory");  // wave-local LDS visibility
    float qp[KK][8];
#pragma unroll
    for (int k = 0; k < KK; k++)
#pragma unroll
      for (int v = 0; v < 8; v++) qp[k][v] = 0.f;
#pragma unroll
    for (int k = 0; k < KK; k++) {
      const float* ck = cinvs + k * DI * DP;
      const float* mk = mus + k * DP;
#pragma unroll
      for (int j = 0; j < 5; j++) {
        v8f acc = {};
        for (int dt = 0; dt < DI / 4; dt++) {
          int d0 = dt * 4 + 2 * half;
          v2f a, b;
          a.x = zw[c * DP + d0]     - mk[d0];
          a.y = zw[c * DP + d0 + 1] - mk[d0 + 1];
          b.x = ck[d0 * DP + j * 16 + c];
          b.y = ck[(d0 + 1) * DP + j * 16 + c];
          acc = __builtin_amdgcn_wmma_f32_16x16x4_f32(
              false, a, false, b, (short)0, acc, false, false);
        }
        float mue = mk[j * 16 + c];
#pragma unroll
        for (int v = 0; v < 8; v++) {
          int m = v + 8 * half;
          qp[k][v] += acc[v] * (zw[m * DP + j * 16 + c] - mue);
        }
      }
    }
    // reduce over the 16 e-lanes of each half
#pragma unroll
    for (int k = 0; k < KK; k++)
#pragma unroll
      for (int v = 0; v < 8; v++) {
        float q = qp[k][v];
        q += __shfl_xor(q, 1); q += __shfl_xor(q, 2);
        q += __shfl_xor(q, 4); q += __shfl_xor(q, 8);
        qp[k][v] = q;
      }
    if (c == 0) {                        // lanes 0 and 16 publish 8 rows each
#pragma unroll
      for (int v = 0; v < 8; v++)
#pragma unroll
        for (int k = 0; k < KK; k++)
          qls[(wave * 16 + v + 8 * half) * KK + k] = qp[k][v];
    }
    asm volatile("s_wait_dscnt 0" ::: "memory");
    if (lane < 16) {
      float tq[KK], mx = 0.f;            // max(0, max_k t) as in reference
#pragma unroll
      for (int k = 0; k < KK; k++) {
        tq[k] = -0.5f * qls[(wave * 16 + lane) * KK + k];
        mx = fmaxf(mx, tq[k]);
      }
      float ssum = 0.f;
#pragma unroll
      for (int k = 0; k < KK; k++) ssum += cf[k] * expf(tq[k] - mx);
      esum += -mx - logf(ssum + EPSC);
    }
  }
  esum += __shfl_xor(esum, 1); esum += __shfl_xor(esum, 2);
  esum += __shfl_xor(esum, 4); esum += __shfl_xor(esum, 8);
  esum += __shfl_xor(esum, 16);
  if (lane == 0) atomicAdd(&ws[OFF_EN], esum);
}

// ---------------------------------------------------------------------------
__global__ void k_final(const float* __restrict__ ws, float* __restrict__ out) {
  if (threadIdx.x == 0 && blockIdx.x == 0) {
    out[0] = ws[OFF_EN] / (float)NS;   // mean sample energy
    out[1] = ws[OFF_CDIAG];            // sum 1/diag(cov)
  }
}

// ---------------------------------------------------------------------------
extern "C" void kernel_launch(void* const* d_in, const int* in_sizes, int n_in,
                              void* d_out, int out_size, void* d_ws, size_t ws_size,
                              hipStream_t stream) {
  (void)in_sizes; (void)n_in; (void)out_size;
  const float* z = (const float*)d_in[0];   // [N, 66] f32
  const float* g = (const float*)d_in[1];   // [N, 4]  f32
  float* ws = (float*)d_ws;
  float* out = (float*)d_out;
  if (ws_size < (size_t)WS_FLOATS * sizeof(float)) return;  // ~192 KB needed

  k_zero  <<<(WS_FLOATS + 255) / 256, 256, 0, stream>>>(ws);
  k_pass1 <<<1024, 256, 0, stream>>>(z, g, ws);
  k_mu    <<<1, 320, 0, stream>>>(ws);
  k_cov   <<<512, 640, 0, stream>>>(z, g, ws);
  k_chol  <<<KK, 128, 0, stream>>>(ws);
  k_energy<<<512, 256, (size_t)LDS3_FLOATS * sizeof(float), stream>>>(z, ws);
  k_final <<<1, 1, 0, stream>>>(ws, out);
}